// GATConv_9242769621446
// MI455X (gfx1250) — compile-verified
//
#include <hip/hip_runtime.h>
#include <hip/hip_bf16.h>
#include <stdint.h>

// ---------------- problem constants (match reference) ----------------
#define NN      8192          // nodes
#define INF_DIM 512           // in_feats
#define NH      8             // heads
#define ND      64            // out feats per head
#define HD      (NH * ND)     // 512
#define NE      262144        // edges
#define TOPK    10
#define NEG_INF (-9e15f)
#define TSZ     (1u << 20)    // hash table slots (load factor 0.25)
#define TMASK   (TSZ - 1u)
#define EMPTY_KEY 0xFFFFFFFFu
#define A_STRIDE 516          // 512 + 4 pad floats -> conflict-free LDS banks

typedef __attribute__((ext_vector_type(2))) float v2f;
typedef __attribute__((ext_vector_type(8))) float v8f;
typedef __attribute__((ext_vector_type(4))) int   v4i;

// ---- CDNA5 async global->LDS availability probes ----
#if defined(__has_builtin)
#if __has_builtin(__builtin_amdgcn_global_load_async_to_lds_b128)
#define HAVE_ASYNC_LDS 1
#endif
#if __has_builtin(__builtin_amdgcn_s_wait_asynccnt)
#define HAVE_WAIT_ASYNC 1
#endif
#endif

// builtin signature (from hipcc diagnostic): param0 = v4i in AS1 (global),
// param1 = v4i in AS3 (LDS), then i32 offset, i32 cpol.
typedef __attribute__((address_space(1))) v4i* g4p;
typedef __attribute__((address_space(3))) v4i* l4p;

// ---------------- helpers ----------------
__device__ __forceinline__ int orderFloat(float f) {
    int i = __float_as_int(f);
    return i >= 0 ? i : (i ^ 0x7FFFFFFF);
}
__device__ __forceinline__ float unorderFloat(int i) {
    return __int_as_float(i >= 0 ? i : (i ^ 0x7FFFFFFF));
}
__device__ __forceinline__ int waveSumI(int v) {
    #pragma unroll
    for (int o = 16; o; o >>= 1) v += __shfl_xor(v, o, 32);
    return v;
}
__device__ __forceinline__ float waveSumF(float v) {
    #pragma unroll
    for (int o = 16; o; o >>= 1) v += __shfl_xor(v, o, 32);
    return v;
}
__device__ __forceinline__ unsigned edgeHash(unsigned key) {
    return (key * 2654435761u) >> 12;   // 20-bit slot seed
}

// ---------------- K1: feat_src = feat @ W^T via V_WMMA_F32_16X16X4_F32 ----------
// Block = 8 waves, all sharing one 16-row M tile. The 16x512 A tile is staged
// once into LDS with GLOBAL_LOAD_ASYNC_TO_LDS_B128 (ASYNCcnt-tracked), then
// each wave computes a 16(M) x 64(N) patch: 4 f32 WMMA accumulators share each
// A fragment.
// A frag (16x4 f32): lane -> (m = lane&15, k = 2*(lane>>4)+v), read from LDS
// (row stride 516 floats -> ds_load_b64 touches all 64 banks exactly once).
// B frag (4x16 f32): same lane mapping with n = lane&15 (B[k][n] = W[n][k]).
__global__ __launch_bounds__(256) void k_gemm_fc(const float* __restrict__ feat,
                                                 const float* __restrict__ W,
                                                 float* __restrict__ fs) {
    __shared__ float Atile[16 * A_STRIDE];           // 33 KB of 320 KB WGP LDS

    const int lane = threadIdx.x & 31;
    const int w    = threadIdx.x >> 5;               // wave id: 0..7 = N group
    const int mt   = blockIdx.x;                     // 0..511 M tile
    const int m0   = mt * 16, n0 = w * 64;
    const int lm   = lane & 15;
    const int ko   = (lane >> 4) << 1;

    // ---- stage A tile: thread t copies 32 floats (row t>>4, segment t&15) ----
    {
        const int row = threadIdx.x >> 4;
        const int seg = threadIdx.x & 15;
        const float* gsrc = feat + (size_t)(m0 + row) * INF_DIM + seg * 32;
        float*       ldst = &Atile[row * A_STRIDE + seg * 32];
#if HAVE_ASYNC_LDS
        #pragma unroll
        for (int j = 0; j < 8; j++) {
            g4p g = (g4p)(uintptr_t)(gsrc + j * 4);
            l4p l = (l4p)(unsigned)(uintptr_t)(ldst + j * 4);
            __builtin_amdgcn_global_load_async_to_lds_b128(g, l, 0, 0);
        }
#if HAVE_WAIT_ASYNC
        __builtin_amdgcn_s_wait_asynccnt(0);
#else
        asm volatile("s_wait_asynccnt 0x0" ::: "memory");
#endif
#else
        #pragma unroll
        for (int j = 0; j < 8; j++)
            *(float4*)(ldst + j * 4) = *(const float4*)(gsrc + j * 4);
#endif
    }
    __syncthreads();

    const float* arow = &Atile[lm * A_STRIDE + ko];              // LDS
    const float* brow = W + (size_t)(n0 + lm) * INF_DIM + ko;    // global

    v8f acc0 = {0.f,0.f,0.f,0.f,0.f,0.f,0.f,0.f};
    v8f acc1 = acc0, acc2 = acc0, acc3 = acc0;

    for (int k = 0; k < INF_DIM; k += 4) {
        v2f a  = *(const v2f*)(arow + k);                        // ds_load_b64
        v2f b0 = *(const v2f*)(brow + k);
        v2f b1 = *(const v2f*)(brow + 16 * INF_DIM + k);
        v2f b2 = *(const v2f*)(brow + 32 * INF_DIM + k);
        v2f b3 = *(const v2f*)(brow + 48 * INF_DIM + k);
        acc0 = __builtin_amdgcn_wmma_f32_16x16x4_f32(false, a, false, b0, (short)0, acc0, false, false);
        acc1 = __builtin_amdgcn_wmma_f32_16x16x4_f32(false, a, false, b1, (short)0, acc1, false, false);
        acc2 = __builtin_amdgcn_wmma_f32_16x16x4_f32(false, a, false, b2, (short)0, acc2, false, false);
        acc3 = __builtin_amdgcn_wmma_f32_16x16x4_f32(false, a, false, b3, (short)0, acc3, false, false);
    }

    // C/D layout: VGPR r -> M = r + 8*(lane>>4), N = lane&15
    const int mrow = m0 + (lane >> 4) * 8;
    #pragma unroll
    for (int r = 0; r < 8; r++) {
        size_t base = (size_t)(mrow + r) * HD + n0 + lm;
        fs[base +  0] = acc0[r];
        fs[base + 16] = acc1[r];
        fs[base + 32] = acc2[r];
        fs[base + 48] = acc3[r];
    }
}

// ---------------- K2: SDDMM — wave per edge, per-head dots + edge val ----------
__global__ __launch_bounds__(256) void k_sddmm(const float* __restrict__ fs,
                                               const int* __restrict__ src,
                                               const int* __restrict__ dst,
                                               const float* __restrict__ trans,
                                               float* __restrict__ ee,
                                               float* __restrict__ val) {
    const int lane = threadIdx.x & 31;
    const int e = blockIdx.x * 8 + (threadIdx.x >> 5);
    if (e >= NE) return;
    const int s = src[e], d = dst[e];
    const float* ps = fs + (size_t)s * HD;
    const float* pd = fs + (size_t)d * HD;
    __builtin_prefetch(ps + lane, 0, 0);
    __builtin_prefetch(pd + lane, 0, 0);
    float p[NH];
    #pragma unroll
    for (int h = 0; h < NH; h++) p[h] = 0.f;
    #pragma unroll
    for (int c = 0; c < 16; c++) {               // idx = c*32+lane, head = c/2
        float x = ps[c * 32 + lane];
        float y = pd[c * 32 + lane];
        p[c >> 1] = fmaf(x, y, p[c >> 1]);
    }
    float tot = 0.f;
    #pragma unroll
    for (int h = 0; h < NH; h++) {
        float r = waveSumF(p[h]);
        if (lane == 0) ee[(size_t)e * NH + h] = r;
        tot += r;
    }
    if (lane == 0) val[e] = tot * trans[e];
}

// ---------------- K3: hash insert of per-edge vals, accumulate duplicates ------
__global__ __launch_bounds__(256) void k_hash_insert(const int* __restrict__ src,
                                                     const int* __restrict__ dst,
                                                     const float* __restrict__ val,
                                                     unsigned* __restrict__ keys,
                                                     float* __restrict__ tvals,
                                                     unsigned* __restrict__ eslot) {
    const int e = blockIdx.x * blockDim.x + threadIdx.x;
    if (e >= NE) return;
    const unsigned key = ((unsigned)src[e] << 13) | (unsigned)dst[e];
    unsigned p = edgeHash(key);
    unsigned slot = 0; unsigned owner = 0;
    for (;;) {
        slot = p & TMASK;
        unsigned prev = atomicCAS(&keys[slot], EMPTY_KEY, key);
        if (prev == EMPTY_KEY) { owner = 0x80000000u; break; }
        if (prev == key) break;
        p++;
    }
    atomicAdd(&tvals[slot], val[e]);
    eslot[e] = slot | owner;
}

// ---------------- K4: CSR build (histogram / scan / scatter / sort) ------------
__global__ __launch_bounds__(256) void k_histogram(const int* __restrict__ src,
                                                   const int* __restrict__ dst,
                                                   int* __restrict__ deg_s,
                                                   int* __restrict__ deg_d) {
    const int e = blockIdx.x * blockDim.x + threadIdx.x;
    if (e >= NE) return;
    atomicAdd(&deg_s[src[e]], 1);
    atomicAdd(&deg_d[dst[e]], 1);
}

__global__ __launch_bounds__(1024) void k_scan8192(const int* __restrict__ in,
                                                   int* __restrict__ out) {
    __shared__ int sh[1024];
    const int t = threadIdx.x;
    int loc[8]; int s = 0;
    #pragma unroll
    for (int i = 0; i < 8; i++) { loc[i] = in[t * 8 + i]; s += loc[i]; }
    sh[t] = s;
    __syncthreads();
    for (int o = 1; o < 1024; o <<= 1) {
        int v = (t >= o) ? sh[t - o] : 0;
        __syncthreads();
        sh[t] += v;
        __syncthreads();
    }
    int run = sh[t] - s;                 // exclusive base
    #pragma unroll
    for (int i = 0; i < 8; i++) { out[t * 8 + i] = run; run += loc[i]; }
    if (t == 1023) out[8192] = sh[1023];
}

__global__ __launch_bounds__(256) void k_scatter(const int* __restrict__ src,
                                                 const int* __restrict__ dst,
                                                 int* __restrict__ cur_s,
                                                 int* __restrict__ cur_d,
                                                 int* __restrict__ csr_s,
                                                 int* __restrict__ csr_d) {
    const int e = blockIdx.x * blockDim.x + threadIdx.x;
    if (e >= NE) return;
    csr_s[atomicAdd(&cur_s[src[e]], 1)] = e;
    csr_d[atomicAdd(&cur_d[dst[e]], 1)] = e;
}

// Insertion-sort each CSR segment by edge id -> deterministic reduction order.
__global__ __launch_bounds__(256) void k_sort_segments(const int* __restrict__ off,
                                                       int* __restrict__ idx) {
    const int r = blockIdx.x * blockDim.x + threadIdx.x;
    if (r >= NN) return;
    const int beg = off[r], end = off[r + 1];
    for (int i = beg + 1; i < end; i++) {
        int key = idx[i]; int j = i - 1;
        while (j >= beg && idx[j] > key) { idx[j + 1] = idx[j]; j--; }
        idx[j + 1] = key;
    }
}

// ---------------- K5: per-row top-10 positive threshold (wave per row) ---------
__global__ __launch_bounds__(256) void k_topk_thresh(const int* __restrict__ offs,
                                                     const int* __restrict__ csrs,
                                                     const unsigned* __restrict__ eslot,
                                                     const float* __restrict__ tvals,
                                                     float* __restrict__ thresh) {
    const int lane = threadIdx.x & 31;
    const int r = blockIdx.x * 8 + (threadIdx.x >> 5);
    if (r >= NN) return;
    const int beg = offs[r], end = offs[r + 1];

    int cnt = 0;
    for (int i = beg + lane; i < end; i += 32) {
        unsigned si = eslot[csrs[i]];
        if (si & 0x80000000u) {
            float u = tvals[si & TMASK];
            if (u > 0.f) cnt++;
        }
    }
    cnt = waveSumI(cnt);

    float t = 0.f;
    if (cnt >= TOPK) {
        for (int j = beg; j < end; j++) {
            unsigned sj = eslot[csrs[j]];
            if (!(sj & 0x80000000u)) continue;
            float v = tvals[sj & TMASK];
            if (v <= 0.f) continue;
            int cgt = 0, cge = 0;
            for (int i = beg + lane; i < end; i += 32) {
                unsigned si = eslot[csrs[i]];
                if (si & 0x80000000u) {
                    float u = tvals[si & TMASK];
                    cgt += (u > v);
                    cge += (u >= v);
                }
            }
            cgt = waveSumI(cgt);
            cge = waveSumI(cge);
            if (cgt <= TOPK - 1 && cge >= TOPK) { t = v; break; }
        }
    }
    if (lane == 0) thresh[r] = t;
}

// ---------------- K6: fill emax with INT_MIN (ordered domain) ------------------
__global__ __launch_bounds__(256) void k_fill_i32(int* __restrict__ p, int v, int n) {
    int i = blockIdx.x * blockDim.x + threadIdx.x;
    if (i < n) p[i] = v;
}

// ---------------- K7: reverse-edge mask + segment max (ordered-int atomicMax) --
__global__ __launch_bounds__(256) void k_mask_and_max(const int* __restrict__ src,
                                                      const int* __restrict__ dst,
                                                      const unsigned* __restrict__ keys,
                                                      const float* __restrict__ tvals,
                                                      const float* __restrict__ thresh,
                                                      float* __restrict__ ee,
                                                      int* __restrict__ emax_i) {
    const int e = blockIdx.x * blockDim.x + threadIdx.x;
    if (e >= NE) return;
    const int s = src[e], d = dst[e];
    // attn_dense[d, s] from the hash table (0 if reverse cell absent)
    const unsigned key = ((unsigned)d << 13) | (unsigned)s;
    unsigned p = edgeHash(key);
    float arev = 0.f;
    for (;;) {
        unsigned slot = p & TMASK;
        unsigned kk = keys[slot];
        if (kk == EMPTY_KEY) break;
        if (kk == key) { arev = tvals[slot]; break; }
        p++;
    }
    const bool cond = arev < thresh[d];     // masked[d,s] < -100
    #pragma unroll
    for (int h = 0; h < NH; h++) {
        float ev = ee[(size_t)e * NH + h];
        if (cond) { ev = NEG_INF; ee[(size_t)e * NH + h] = ev; }
        atomicMax(&emax_i[d * NH + h], orderFloat(ev));
    }
}

// ---------------- K8: e_exp = exp(e - emax[dst]) (in place) --------------------
__global__ __launch_bounds__(256) void k_edge_exp(const int* __restrict__ dst,
                                                  const int* __restrict__ emax_i,
                                                  float* __restrict__ ee) {
    const int e = blockIdx.x * blockDim.x + threadIdx.x;
    if (e >= NE) return;
    const int d = dst[e];
    #pragma unroll
    for (int h = 0; h < NH; h++) {
        float m = unorderFloat(emax_i[d * NH + h]);
        ee[(size_t)e * NH + h] = expf(ee[(size_t)e * NH + h] - m);
    }
}

// ---------------- K9: deterministic segment sums over sorted CSR ---------------
__global__ __launch_bounds__(256) void k_seg_esum(const float* __restrict__ ee,
                                                  const int* __restrict__ offd,
                                                  const int* __restrict__ csrd,
                                                  float* __restrict__ esum) {
    const int i = blockIdx.x * blockDim.x + threadIdx.x;     // (v, h)
    if (i >= NN * NH) return;
    const int v = i >> 3;
    const int h = i & 7;
    const int beg = offd[v], end = offd[v + 1];
    float s = 0.f;
    for (int j = beg; j < end; j++) s += ee[(size_t)csrd[j] * NH + h];
    esum[i] = s;
}

// ---------------- K10: aggregation — workgroup per dst node, no atomics --------
__global__ __launch_bounds__(128) void k_aggregate(const float* __restrict__ fs,
                                                   const float* __restrict__ ee,
                                                   const float* __restrict__ esum,
                                                   const int* __restrict__ offd,
                                                   const int* __restrict__ csrd,
                                                   const int* __restrict__ src,
                                                   float* __restrict__ out) {
    const int v = blockIdx.x;
    const int t = threadIdx.x;             // 0..127, cols 4t..4t+3
    const int h = t >> 4;                  // single head per thread
    const int beg = offd[v], end = offd[v + 1];
    float a0 = 0.f, a1 = 0.f, a2 = 0.f, a3 = 0.f;
    for (int i = beg; i < end; i++) {
        const int e = csrd[i];
        const float* row = fs + (size_t)src[e] * HD + t * 4;
        if (i + 1 < end)
            __builtin_prefetch(fs + (size_t)src[csrd[i + 1]] * HD + t * 4, 0, 0);
        const float w = ee[(size_t)e * NH + h];
        const float4 f = *(const float4*)row;
        a0 = fmaf(w, f.x, a0);
        a1 = fmaf(w, f.y, a1);
        a2 = fmaf(w, f.z, a2);
        a3 = fmaf(w, f.w, a3);
    }
    const float es = esum[v * NH + h];
    const float inv = es > 0.f ? 1.0f / es : 0.f;
    float4 r; r.x = a0 * inv; r.y = a1 * inv; r.z = a2 * inv; r.w = a3 * inv;
    *(float4*)(out + (size_t)v * HD + t * 4) = r;
}

// ---------------- host launch ----------------
extern "C" void kernel_launch(void* const* d_in, const int* in_sizes, int n_in,
                              void* d_out, int out_size, void* d_ws, size_t ws_size,
                              hipStream_t stream) {
    (void)in_sizes; (void)n_in; (void)out_size; (void)ws_size;

    const float* feat  = (const float*)d_in[0];
    const float* W     = (const float*)d_in[1];
    const float* trans = (const float*)d_in[2];
    const int*   src   = (const int*)d_in[3];
    const int*   dst   = (const int*)d_in[4];
    float*       out   = (float*)d_out;

    // ---- carve workspace ----
    char* ws = (char*)d_ws;
    size_t off = 0;
    auto carve = [&](size_t bytes) -> void* {
        void* p = ws + off;
        off = (off + bytes + 255) & ~(size_t)255;
        return p;
    };
    float*    fs      = (float*)   carve((size_t)NN * HD * 4);   // 16 MB
    float*    ee      = (float*)   carve((size_t)NE * NH * 4);   //  8 MB
    float*    val     = (float*)   carve((size_t)NE * 4);        //  1 MB
    unsigned* keys    = (unsigned*)carve((size_t)TSZ * 4);       //  4 MB
    float*    tvals   = (float*)   carve((size_t)TSZ * 4);       //  4 MB
    unsigned* eslot   = (unsigned*)carve((size_t)NE * 4);        //  1 MB
    int*      deg_s   = (int*)     carve((size_t)NN * 4);
    int*      deg_d   = (int*)     carve((size_t)NN * 4);
    int*      off_s   = (int*)     carve((size_t)(NN + 1) * 4);
    int*      off_d   = (int*)     carve((size_t)(NN + 1) * 4);
    int*      cur_s   = (int*)     carve((size_t)NN * 4);
    int*      cur_d   = (int*)     carve((size_t)NN * 4);
    int*      csr_s   = (int*)     carve((size_t)NE * 4);        //  1 MB
    int*      csr_d   = (int*)     carve((size_t)NE * 4);        //  1 MB
    float*    thresh  = (float*)   carve((size_t)NN * 4);
    int*      emax_i  = (int*)     carve((size_t)NN * NH * 4);
    float*    esum    = (float*)   carve((size_t)NN * NH * 4);

    // ---- init (capture-safe async ops only) ----
    (void)hipMemsetAsync(keys,  0xFF, (size_t)TSZ * 4, stream);
    (void)hipMemsetAsync(tvals, 0x00, (size_t)TSZ * 4, stream);
    (void)hipMemsetAsync(deg_s, 0x00, (size_t)NN * 4, stream);
    (void)hipMemsetAsync(deg_d, 0x00, (size_t)NN * 4, stream);

    // ---- pipeline ----
    k_gemm_fc<<<512, 256, 0, stream>>>(feat, W, fs);
    k_sddmm<<<NE / 8, 256, 0, stream>>>(fs, src, dst, trans, ee, val);
    k_hash_insert<<<NE / 256, 256, 0, stream>>>(src, dst, val, keys, tvals, eslot);
    k_histogram<<<NE / 256, 256, 0, stream>>>(src, dst, deg_s, deg_d);
    k_scan8192<<<1, 1024, 0, stream>>>(deg_s, off_s);
    k_scan8192<<<1, 1024, 0, stream>>>(deg_d, off_d);
    (void)hipMemcpyAsync(cur_s, off_s, (size_t)NN * 4, hipMemcpyDeviceToDevice, stream);
    (void)hipMemcpyAsync(cur_d, off_d, (size_t)NN * 4, hipMemcpyDeviceToDevice, stream);
    k_scatter<<<NE / 256, 256, 0, stream>>>(src, dst, cur_s, cur_d, csr_s, csr_d);
    k_sort_segments<<<NN / 256, 256, 0, stream>>>(off_s, csr_s);
    k_sort_segments<<<NN / 256, 256, 0, stream>>>(off_d, csr_d);
    k_fill_i32<<<(NN * NH) / 256, 256, 0, stream>>>(emax_i, INT32_MIN, NN * NH);
    k_topk_thresh<<<NN / 8, 256, 0, stream>>>(off_s, csr_s, eslot, tvals, thresh);
    k_mask_and_max<<<NE / 256, 256, 0, stream>>>(src, dst, keys, tvals, thresh, ee, emax_i);
    k_edge_exp<<<NE / 256, 256, 0, stream>>>(dst, emax_i, ee);
    k_seg_esum<<<(NN * NH) / 256, 256, 0, stream>>>(ee, off_d, csr_d, esum);
    k_aggregate<<<NN, 128, 0, stream>>>(fs, ee, esum, off_d, csr_d, src, out);
}